// GraphConvolution_68908455297307
// MI455X (gfx1250) — compile-verified
//
#include <hip/hip_runtime.h>

// GraphConv for MI455X (gfx1250, wave32):
//   agg = features            (self loop)
//   agg[dst[e]] += features[src[e]]   for all edges   (global_atomic_add_f32)
//   out = agg @ W + bias      (V_WMMA_F32_16X16X4_F32, exact fp32)

typedef float v2f __attribute__((ext_vector_type(2)));
typedef float v4f __attribute__((ext_vector_type(4)));
typedef float v8f __attribute__((ext_vector_type(8)));

#define IN_FEAT  128
#define OUT_FEAT 256

// ---------------------------------------------------------------------------
// Kernel 1: agg = features  (vectorized 16B copy; fully overwrites workspace
// every call so the 0xAA poison never survives)
// ---------------------------------------------------------------------------
__global__ __launch_bounds__(256) void gc_init_agg(const v4f* __restrict__ feat,
                                                   v4f* __restrict__ agg, int n4) {
  int i = blockIdx.x * blockDim.x + threadIdx.x;
  if (i < n4) agg[i] = feat[i];
}

// ---------------------------------------------------------------------------
// Kernel 2: edge scatter-add. One wave (32 lanes) per edge; each lane moves a
// 16B chunk of the 512B feature row and issues 4 native f32 atomics. The
// 51.2MB accumulator is L2-resident (192MB L2), so these resolve in the L2
// atomic units. Inline asm guarantees global_atomic_add_f32 (no CAS loop).
// ---------------------------------------------------------------------------
__global__ __launch_bounds__(256) void gc_edge_scatter(const float* __restrict__ feat,
                                                       const int* __restrict__ src,
                                                       const int* __restrict__ dst,
                                                       float* __restrict__ agg,
                                                       int n_edges) {
  long long t = (long long)blockIdx.x * blockDim.x + threadIdx.x;
  int e    = (int)(t >> 5);
  int lane = (int)(t & 31);
  if (e >= n_edges) return;  // wave-uniform exit

  int s = src[e];
  int d = dst[e];
  v4f v = *(const v4f*)(feat + (long long)s * IN_FEAT + lane * 4);
  float* p = agg + (long long)d * IN_FEAT + lane * 4;

  // STOREcnt is drained by the implicit wait-idle at s_endpgm.
  asm volatile("global_atomic_add_f32 %0, %1, off\n\t"
               "global_atomic_add_f32 %0, %2, off offset:4\n\t"
               "global_atomic_add_f32 %0, %3, off offset:8\n\t"
               "global_atomic_add_f32 %0, %4, off offset:12"
               :
               : "v"(p), "v"(v.x), "v"(v.y), "v"(v.z), "v"(v.w)
               : "memory");
}

// ---------------------------------------------------------------------------
// Kernel 3: out[M,256] = agg[M,128] @ W[128,256] + bias, fp32 WMMA.
// One wave per 16x16 output tile; 32x V_WMMA_F32_16X16X4_F32 over K=128.
//
// ISA 7.12.2 layouts (32-bit, wave32):
//   A 16x4 : lane%16 = M, lane/16 = K-half, VGPR pair = {K, K+1}  -> float2 load
//   B 4x16 : lane%16 = N, lane/16 = K-half, VGPR pair = {K, K+1}  -> 2 scalar loads
//   C 16x16: VGPR i holds row i + 8*(lane/16), col = lane%16
// Requires M % 16 == 0 (EXEC must be all ones; 100000 % 16 == 0 here).
// ---------------------------------------------------------------------------
__global__ __launch_bounds__(256) void gc_gemm_wmma(const float* __restrict__ h,
                                                    const float* __restrict__ W,
                                                    const float* __restrict__ bias,
                                                    float* __restrict__ out,
                                                    int m_tiles) {
  const int NT = OUT_FEAT / 16;  // 16 N-tiles
  int wave = (int)((blockIdx.x * (long long)blockDim.x + threadIdx.x) >> 5);
  int lane = threadIdx.x & 31;
  int mt = wave / NT;
  int nt = wave % NT;
  if (mt >= m_tiles) return;  // wave-uniform: EXEC stays all-ones for WMMA

  int row = lane & 15;  // M index for A, N index for B/D
  int kh  = lane >> 4;  // K-half select (0 or 1)

  const float* ha = h + (long long)(mt * 16 + row) * IN_FEAT + kh * 2;
  const float* wb = W + (long long)(kh * 2) * OUT_FEAT + nt * 16 + row;

  v8f c = {};
#pragma unroll
  for (int k0 = 0; k0 < IN_FEAT; k0 += 4) {
    v2f a = *(const v2f*)(ha + k0);              // A(m, k0+2kh .. +1), 8B aligned
    v2f b;
    b.x = wb[(long long)k0 * OUT_FEAT];          // B(k0+2kh,   n)
    b.y = wb[(long long)(k0 + 1) * OUT_FEAT];    // B(k0+2kh+1, n)
    // (neg_a, A, neg_b, B, c_mod, C, reuse_a, reuse_b)
    c = __builtin_amdgcn_wmma_f32_16x16x4_f32(false, a, false, b,
                                              (short)0, c, false, false);
  }

  float bv = bias[nt * 16 + row];
  long long obase = (long long)(mt * 16 + kh * 8) * OUT_FEAT + nt * 16 + row;
#pragma unroll
  for (int i = 0; i < 8; ++i)
    out[obase + (long long)i * OUT_FEAT] = c[i] + bv;
}

// ---------------------------------------------------------------------------
extern "C" void kernel_launch(void* const* d_in, const int* in_sizes, int n_in,
                              void* d_out, int out_size, void* d_ws, size_t ws_size,
                              hipStream_t stream) {
  const float* feat = (const float*)d_in[0];   // [N, 128] f32
  const int*   src  = (const int*)d_in[1];     // [E] i32
  const int*   dst  = (const int*)d_in[2];     // [E] i32
  const float* W    = (const float*)d_in[3];   // [128, 256] f32
  const float* bias = (const float*)d_in[4];   // [256] f32
  float*       out  = (float*)d_out;           // [N, 256] f32
  float*       agg  = (float*)d_ws;            // [N, 128] f32 scratch (51.2 MB)

  int n_feat  = in_sizes[0];
  int n_nodes = n_feat / IN_FEAT;
  int n_edges = in_sizes[1];

  // 1) agg = features (self loop)
  int n4 = n_feat / 4;
  gc_init_agg<<<(n4 + 255) / 256, 256, 0, stream>>>((const v4f*)feat, (v4f*)agg, n4);

  // 2) scatter-add over edges: one wave per edge
  long long t2 = (long long)n_edges * 32;
  int blocks2 = (int)((t2 + 255) / 256);
  gc_edge_scatter<<<blocks2, 256, 0, stream>>>(feat, src, dst, agg, n_edges);

  // 3) fp32 WMMA GEMM + bias: one wave per 16x16 output tile (8 waves/block)
  int m_tiles = n_nodes / 16;                  // 100000/16 = 6250, exact
  int waves   = m_tiles * (OUT_FEAT / 16);     // 100000 waves
  gc_gemm_wmma<<<(waves + 7) / 8, 256, 0, stream>>>(agg, W, bias, out, m_tiles);
}